// BBoxGenerator_26259430047833
// MI455X (gfx1250) — compile-verified
//
#include <hip/hip_runtime.h>

#define HH 512
#define WW 512
#define NTHREADS 1024
#define SPLIT 4                       // blocks per image
#define STAGES_TOTAL 64               // 512*512/4 float4s / 1024 threads
#define STAGES_PER (STAGES_TOTAL / SPLIT)
#define STAGE_ELTS (NTHREADS * 4)     // 4096 elements per stage per block
#define NBUF 3                        // triple buffer: no LDS reuse race window
#define HALF_BITS 0x3F000000          // bits of 0.5f; int order == float order for x>=0

typedef int v4i __attribute__((ext_vector_type(4)));
typedef __attribute__((address_space(1))) v4i gv4i; // global
typedef __attribute__((address_space(3))) v4i lv4i; // LDS

#if __has_builtin(__builtin_amdgcn_global_load_async_to_lds_b128)
#define HAVE_ASYNC_LDS 1
#else
#define HAVE_ASYNC_LDS 0
#endif

#if __has_builtin(__builtin_amdgcn_s_wait_asynccnt)
#define WAIT_ASYNC(n) __builtin_amdgcn_s_wait_asynccnt(n)
#else
#define WAIT_ASYNC(n) asm volatile("s_wait_asynccnt %0" ::"i"(n) : "memory")
#endif

__device__ __forceinline__ int imin(int a, int b) { return a < b ? a : b; }
__device__ __forceinline__ int imax(int a, int b) { return a > b ? a : b; }

#if HAVE_ASYNC_LDS
__device__ __forceinline__ void async_copy16(const int* g, int* l) {
  // global (AS1) -> LDS (AS3), 16 bytes per lane, tracked by ASYNCcnt
  __builtin_amdgcn_global_load_async_to_lds_b128((gv4i*)g, (lv4i*)l, 0, 0);
}
#endif

// Pass 0: write min/max identities into d_out (as ints)
__global__ void bbox_init(int* __restrict__ oi, int nimg) {
  int i = blockIdx.x * blockDim.x + threadIdx.x;
  if (i < nimg) {
    oi[i * 4 + 0] = WW;  // xmin identity
    oi[i * 4 + 1] = -1;  // xmax identity
    oi[i * 4 + 2] = HH;  // ymin identity
    oi[i * 4 + 3] = -1;  // ymax identity
  }
}

// Pass 1: each block streams 1/SPLIT of one image, reduces, atomically merges.
__global__ __launch_bounds__(NTHREADS) void bbox_partial(
    const float* __restrict__ mask, int* __restrict__ oi) {
  const int blk = blockIdx.x;
  const int b = blk / SPLIT;
  const int p = blk % SPLIT;
  const int tid = threadIdx.x;
  const int* img = (const int*)(mask + (size_t)b * (HH * WW));

  // Thread's float4 covers x0..x0+3 of row y = s*8 + ybase at global stage s.
  const int x0 = (tid & 127) << 2;
  const int ybase = tid >> 7;
  const int s0 = p * STAGES_PER;
  const int* gptr = img + (size_t)s0 * STAGE_ELTS + tid * 4;

  // Integer max accumulators over raw f32 bits (valid: mask values >= 0).
  int a0 = 0, a1 = 0, a2 = 0, a3 = 0;
  int ymin = HH, ymax = -1;

#define PROC(w, yy)                                                  \
  do {                                                               \
    int _m = imax(imax((w).x, (w).y), imax((w).z, (w).w));           \
    if (_m > HALF_BITS) { ymin = imin(ymin, (yy)); ymax = (yy); }    \
    a0 = imax(a0, (w).x); a1 = imax(a1, (w).y);                      \
    a2 = imax(a2, (w).z); a3 = imax(a3, (w).w);                      \
  } while (0)

#if HAVE_ASYNC_LDS
  __shared__ __align__(16) int lbuf[NBUF][STAGE_ELTS]; // 48 KB triple buffer
  async_copy16(gptr, &lbuf[0][tid << 2]);
  int cur = 0, nxt = 1;
  for (int ls = 0; ls < STAGES_PER; ++ls) {
    if (ls + 1 < STAGES_PER) {
      async_copy16(gptr + (size_t)(ls + 1) * STAGE_ELTS, &lbuf[nxt][tid << 2]);
      WAIT_ASYNC(1); // stage ls complete (in-order), stage ls+1 in flight
    } else {
      WAIT_ASYNC(0);
    }
    v4i w = *(const v4i*)&lbuf[cur][tid << 2];
    PROC(w, (s0 + ls) * 8 + ybase);
    cur = nxt;
    nxt = (nxt + 1 == NBUF) ? 0 : nxt + 1;
  }
#else
#pragma unroll 4
  for (int ls = 0; ls < STAGES_PER; ++ls) {
    v4i w = *(const v4i*)(gptr + (size_t)ls * STAGE_ELTS);
    PROC(w, (s0 + ls) * 8 + ybase);
  }
#endif
#undef PROC

  // x extents from accumulated per-column maxima (constant x0 per thread).
  bool c0 = a0 > HALF_BITS, c1 = a1 > HALF_BITS;
  bool c2 = a2 > HALF_BITS, c3 = a3 > HALF_BITS;
  bool anyx = c0 | c1 | c2 | c3;
  int first = c0 ? 0 : (c1 ? 1 : (c2 ? 2 : 3));
  int last = c3 ? 3 : (c2 ? 2 : (c1 ? 1 : 0));
  int xmin = anyx ? (x0 + first) : WW;
  int xmax = anyx ? (x0 + last) : -1;

  // wave32 tree reduction
#pragma unroll
  for (int off = 16; off > 0; off >>= 1) {
    xmin = imin(xmin, __shfl_xor(xmin, off, 32));
    xmax = imax(xmax, __shfl_xor(xmax, off, 32));
    ymin = imin(ymin, __shfl_xor(ymin, off, 32));
    ymax = imax(ymax, __shfl_xor(ymax, off, 32));
  }

  __shared__ int rxmin[32], rxmax[32], rymin[32], rymax[32];
  if ((tid & 31) == 0) {
    int w = tid >> 5;
    rxmin[w] = xmin; rxmax[w] = xmax; rymin[w] = ymin; rymax[w] = ymax;
  }
  __syncthreads();
  if (tid < 32) {
    xmin = rxmin[tid]; xmax = rxmax[tid]; ymin = rymin[tid]; ymax = rymax[tid];
#pragma unroll
    for (int off = 16; off > 0; off >>= 1) {
      xmin = imin(xmin, __shfl_xor(xmin, off, 32));
      xmax = imax(xmax, __shfl_xor(xmax, off, 32));
      ymin = imin(ymin, __shfl_xor(ymin, off, 32));
      ymax = imax(ymax, __shfl_xor(ymax, off, 32));
    }
    if (tid == 0) {
      atomicMin(&oi[b * 4 + 0], xmin);
      atomicMax(&oi[b * 4 + 1], xmax);
      atomicMin(&oi[b * 4 + 2], ymin);
      atomicMax(&oi[b * 4 + 3], ymax);
    }
  }
}

// Pass 2: convert merged int extents (in d_out) to final float boxes in place.
__global__ void bbox_final(float* __restrict__ out, int nimg) {
  int b = blockIdx.x * blockDim.x + threadIdx.x;
  if (b >= nimg) return;
  const int* oi = (const int*)out;
  int xmin = oi[b * 4 + 0], xmax = oi[b * 4 + 1];
  int ymin = oi[b * 4 + 2], ymax = oi[b * 4 + 3];

  bool empty = (ymax < 0);
  const float s = 0.05f;
  float xlo = (float)xmin * (1.0f / 512.0f);
  float xhi = (float)xmax * (1.0f / 512.0f);
  float ylo = (float)ymin * (1.0f / 512.0f);
  float yhi = (float)ymax * (1.0f / 512.0f);
  {
    bool small = (xhi - xlo) < s;
    float c = (xlo + xhi) * 0.5f;
    float lo2 = fmaxf(0.0f, c - s * 0.5f);
    float hi2 = fminf(1.0f, c + s * 0.5f);
    xlo = small ? lo2 : xlo;
    xhi = small ? hi2 : xhi;
  }
  {
    bool small = (yhi - ylo) < s;
    float c = (ylo + yhi) * 0.5f;
    float lo2 = fmaxf(0.0f, c - s * 0.5f);
    float hi2 = fminf(1.0f, c + s * 0.5f);
    ylo = small ? lo2 : ylo;
    yhi = small ? hi2 : yhi;
  }
  float4 box = empty ? make_float4(0.25f, 0.25f, 0.75f, 0.75f)
                     : make_float4(xlo, ylo, xhi, yhi);
  ((float4*)out)[b] = box;
}

extern "C" void kernel_launch(void* const* d_in, const int* in_sizes, int n_in,
                              void* d_out, int out_size, void* d_ws, size_t ws_size,
                              hipStream_t stream) {
  const float* mask = (const float*)d_in[0];
  const int B = in_sizes[0] / (HH * WW);
  int* oi = (int*)d_out;
  bbox_init<<<dim3((B + 255) / 256), dim3(256), 0, stream>>>(oi, B);
  bbox_partial<<<dim3(B * SPLIT), dim3(NTHREADS), 0, stream>>>(mask, oi);
  bbox_final<<<dim3((B + 255) / 256), dim3(256), 0, stream>>>((float*)d_out, B);
}